// FittedWarp_38027640439463
// MI455X (gfx1250) — compile-verified
//
#include <hip/hip_runtime.h>
#include <math.h>

// Problem constants from the reference: x:[8,2048,256] f32, w:[256,1] f32.
#define B_ 8
#define N_ 2048
#define D_ 256

typedef __attribute__((ext_vector_type(2))) float v2f;
typedef __attribute__((ext_vector_type(8))) float v8f;

// ---------------------------------------------------------------------------
// Kernel 0: sentinel-init the segment-start table: start[b][m] = N for all m
// in [0, N]. start[b][m] will later hold the first column i whose idx == m.
// ---------------------------------------------------------------------------
__global__ void init_start_kernel(int* __restrict__ start) {
  int i = blockIdx.x * blockDim.x + threadIdx.x;
  if (i < B_ * (N_ + 1)) start[i] = N_;
}

// ---------------------------------------------------------------------------
// Kernel 1: a[g] = sigmoid(dot(x[g,:], w)) via V_WMMA_F32_16X16X4_F32.
// One wave (32 threads) handles 16 rows. Per ISA 32-bit A layout (16x4):
//   lanes 0-15:  M=lane,    VGPR0=K0, VGPR1=K1
//   lanes 16-31: M=lane-16, VGPR0=K2, VGPR1=K3
// B (4x16) gets w[k] replicated across all 16 N columns (same float2 per
// half-wave), so every column of the 16x16 f32 accumulator equals the dot.
// Column N=0 lives in lane 0 (M=0..7, VGPR 0..7) and lane 16 (M=8..15).
// ---------------------------------------------------------------------------
__global__ __launch_bounds__(32) void matvec_sigmoid_wmma(
    const float* __restrict__ x, const float* __restrict__ w,
    float* __restrict__ a_out) {
  const int lane = threadIdx.x;                 // 0..31, wave32
  const long row0 = (long)blockIdx.x * 16;      // 16 rows per wave
  const int m = lane & 15;
  const int koff = (lane >> 4) * 2;             // 0 for lanes 0-15, 2 for 16-31
  const float* xrow = x + (row0 + m) * D_;

  v8f c = {};
  #pragma unroll 4
  for (int k0 = 0; k0 < D_; k0 += 4) {
    v2f av = *(const v2f*)(xrow + k0 + koff);   // A: 16x4 f32 tile of x
    v2f bv = *(const v2f*)(w + k0 + koff);      // B: w chunk, replicated in N
    // 8 args: (neg_a, A, neg_b, B, c_mod, C, reuse_a, reuse_b)
    c = __builtin_amdgcn_wmma_f32_16x16x4_f32(
        false, av, false, bv, (short)0, c, false, false);
  }

  if (lane == 0 || lane == 16) {
    const int mbase = (lane == 0) ? 0 : 8;
    #pragma unroll
    for (int r = 0; r < 8; ++r) {
      float v = c[r];                            // (M = mbase+r, N = 0)
      a_out[row0 + mbase + r] = 1.0f / (1.0f + expf(-v));
    }
  }
}

// ---------------------------------------------------------------------------
// Kernel 2: per-batch inclusive scan of a (one wave per batch, 64 elems per
// lane, wave32 shuffle scan of lane totals), then per-column warp weights:
//   idx  = floor(cumsum);  cross = idx != idx_prev;  frac = cumsum - idx
//   main weight  (row idx):   cross ? frac     : a
//   second weight(row idx-1): cross ? a - frac : 0
// Each crossing column is the unique first column of its segment -> plain
// store into start[b][idx], no atomics.
// ---------------------------------------------------------------------------
__global__ __launch_bounds__(32) void scan_kernel(
    const float* __restrict__ a, float* __restrict__ wm,
    float* __restrict__ wsec, int* __restrict__ start) {
  const int b = blockIdx.x;
  const int lane = threadIdx.x;
  const int CH = N_ / 32;                        // 64 elements per lane
  const float* ab = a + b * N_;
  const int i0 = lane * CH;

  float s = 0.0f;
  for (int i = 0; i < CH; ++i) s += ab[i0 + i];

  // wave32 inclusive scan of lane totals
  float v = s;
  for (int off = 1; off < 32; off <<= 1) {
    float n = __shfl_up(v, off, 32);
    if (lane >= off) v += n;
  }
  const float prefix = v - s;                    // exclusive prefix (cumsum through i0-1)

  float run = prefix;
  int prev = (int)floorf(prefix);                // idx of element i0-1 (0 for lane 0)
  int* sb = start + b * (N_ + 1);
  float* wmb = wm + b * N_;
  float* wsb = wsec + b * N_;

  for (int i = 0; i < CH; ++i) {
    const float ai = ab[i0 + i];
    run += ai;
    const int idx = (int)floorf(run);
    const float frac = run - (float)idx;
    const bool cross = (idx != prev);
    wmb[i0 + i] = cross ? frac : ai;
    wsb[i0 + i] = cross ? (ai - frac) : 0.0f;
    if (cross) sb[idx] = i0 + i;                 // unique writer per segment
    prev = idx;
  }
}

// ---------------------------------------------------------------------------
// Kernel 3: out[b,m,:] = sum_{i in segment m} wm[i]*x[b,i,:]
//                      + (segment m+1 exists ? wsec[first(m+1)]*x[b,first,:] : 0)
// One 64-thread block per output row; 4 floats (float4) per thread.
// Rows beyond idx_max get s0==s1==N -> write zeros (no separate memset pass).
// ---------------------------------------------------------------------------
__global__ __launch_bounds__(64) void gather_rows(
    const float* __restrict__ x, const float* __restrict__ wm,
    const float* __restrict__ wsec, const int* __restrict__ start,
    float* __restrict__ out) {
  const int m = blockIdx.x;
  const int b = blockIdx.y;
  const int d0 = threadIdx.x * 4;

  const int* sb = start + b * (N_ + 1);
  const int s0 = (m == 0) ? 0 : sb[m];
  const int s1 = sb[m + 1];                      // <= N (sentinel)

  const float* xb = x + (size_t)b * N_ * D_;
  const float* wmb = wm + b * N_;

  float4 acc = make_float4(0.f, 0.f, 0.f, 0.f);
  for (int i = s0; i < s1; ++i) {
    const float g = wmb[i];
    const float4 xv = *(const float4*)(xb + (size_t)i * D_ + d0);
    acc.x += g * xv.x; acc.y += g * xv.y; acc.z += g * xv.z; acc.w += g * xv.w;
  }
  if (s1 < N_) {                                 // crossing column of segment m+1
    const float g = wsec[b * N_ + s1];
    const float4 xv = *(const float4*)(xb + (size_t)s1 * D_ + d0);
    acc.x += g * xv.x; acc.y += g * xv.y; acc.z += g * xv.z; acc.w += g * xv.w;
  }
  *(float4*)(out + ((size_t)b * N_ + m) * D_ + d0) = acc;
}

// ---------------------------------------------------------------------------
extern "C" void kernel_launch(void* const* d_in, const int* in_sizes, int n_in,
                              void* d_out, int out_size, void* d_ws, size_t ws_size,
                              hipStream_t stream) {
  const float* x = (const float*)d_in[0];        // [B, N, D] f32
  const float* w = (const float*)d_in[1];        // [D, 1]   f32
  float* out = (float*)d_out;                    // [B, N, D] f32

  // Workspace layout (~256 KB total)
  float* a_buf = (float*)d_ws;                   // B*N   f32
  float* wm    = a_buf + B_ * N_;                // B*N   f32
  float* wsec  = wm + B_ * N_;                   // B*N   f32
  int*   start = (int*)(wsec + B_ * N_);         // B*(N+1) i32

  const int nstart = B_ * (N_ + 1);
  init_start_kernel<<<(nstart + 255) / 256, 256, 0, stream>>>(start);

  matvec_sigmoid_wmma<<<(B_ * N_) / 16, 32, 0, stream>>>(x, w, a_buf);

  scan_kernel<<<B_, 32, 0, stream>>>(a_buf, wm, wsec, start);

  gather_rows<<<dim3(N_, B_), 64, 0, stream>>>(x, wm, wsec, start, out);
}